// RelationAwareGCN_14499809591443
// MI455X (gfx1250) — compile-verified
//
#include <hip/hip_runtime.h>
#include <hip/hip_bf16.h>
#include <math.h>

// ---------------------------------------------------------------------------
// RelationAwareGCN for MI455X (gfx1250, wave32).
// GEMMs use V_WMMA_F32_16X16X4_F32 (exact fp32; GEMM flops are negligible vs
// the L2-bound SpMM scatters, so no reason to quantize).
// SpMM = edge-parallel gather + global_atomic_add_f32 into L2 (data resident
// in 192MB L2). Note: float atomics give run-to-run summation-order variance
// (standard for scatter-add SpMM); per-call work is identical.
// ---------------------------------------------------------------------------

typedef __attribute__((ext_vector_type(2))) float v2f;
typedef __attribute__((ext_vector_type(8))) float v8f;

#define HDIM 128
#define BN_EPS 1e-5f

// ---------------- elementwise helpers ----------------

__global__ void zero_kernel(float* __restrict__ p, size_t n) {
    size_t i = (size_t)blockIdx.x * blockDim.x + threadIdx.x;
    size_t stride = (size_t)gridDim.x * blockDim.x;
    for (; i < n; i += stride) p[i] = 0.0f;
}

__global__ void degree_kernel(const int* __restrict__ src, const int* __restrict__ dst,
                              float* __restrict__ in_deg, float* __restrict__ out_deg, int E) {
    int e = blockIdx.x * blockDim.x + threadIdx.x;
    if (e >= E) return;
    atomicAdd(&out_deg[src[e]], 1.0f);
    atomicAdd(&in_deg[dst[e]], 1.0f);
}

__global__ void invdeg_kernel(const float* __restrict__ in_deg, const float* __restrict__ out_deg,
                              float* __restrict__ inv_in, float* __restrict__ inv_out,
                              float* __restrict__ inv_mean, int N) {
    int i = blockIdx.x * blockDim.x + threadIdx.x;
    if (i >= N) return;
    float id = fmaxf(in_deg[i], 1.0f);
    float od = fmaxf(out_deg[i], 1.0f);
    inv_in[i]   = rsqrtf(id);   // norm='both' dst side
    inv_out[i]  = rsqrtf(od);   // norm='both' src side
    inv_mean[i] = 1.0f / id;    // mean aggregation
}

// Edge-parallel scatter-add: one wave handles one edge's full 128-float row
// (32 lanes x float4).  src/dst loads are wave-uniform -> scalar cache.
__global__ void scatter_add_kernel(const float* __restrict__ x,
                                   const float* __restrict__ srcscale, // nullable
                                   const int* __restrict__ src, const int* __restrict__ dst,
                                   float* __restrict__ accum, long long total) {
    long long gid = (long long)blockIdx.x * blockDim.x + threadIdx.x;
    if (gid >= total) return;
    int e = (int)(gid >> 5);
    int f = (int)(gid & 31) << 2;
    int s = src[e];
    int d = dst[e];
    float sc = srcscale ? srcscale[s] : 1.0f;
    const float4 v = *(const float4*)(x + (size_t)s * HDIM + f);
    float* acc = accum + (size_t)d * HDIM + f;
    atomicAdd(acc + 0, v.x * sc);
    atomicAdd(acc + 1, v.y * sc);
    atomicAdd(acc + 2, v.z * sc);
    atomicAdd(acc + 3, v.w * sc);
}

// attn = sigmoid(a @ w2 + b2): one wave per node, shfl_xor reduction.
__global__ void attn_score_kernel(const float* __restrict__ a, const float* __restrict__ w2,
                                  const float* __restrict__ b2, float* __restrict__ attn, int N) {
    int node = blockIdx.x * (blockDim.x >> 5) + (threadIdx.x >> 5);
    if (node >= N) return;
    int lane = threadIdx.x & 31;
    const float4 av = *(const float4*)(a + (size_t)node * HDIM + lane * 4);
    const float4 wv = *(const float4*)(w2 + lane * 4);
    float s = av.x * wv.x + av.y * wv.y + av.z * wv.z + av.w * wv.w;
    #pragma unroll
    for (int off = 16; off > 0; off >>= 1) s += __shfl_xor(s, off, 32);
    if (lane == 0) attn[node] = 1.0f / (1.0f + __expf(-(s + b2[0])));
}

// h_att = h + agg * attn[node]   (in place on h)
__global__ void gated_add_kernel(float* __restrict__ h, const float* __restrict__ agg,
                                 const float* __restrict__ attn, size_t total) {
    size_t i = (size_t)blockIdx.x * blockDim.x + threadIdx.x;
    if (i >= total) return;
    h[i] = h[i] + agg[i] * attn[i >> 7]; // HDIM == 128
}

// ---------------- WMMA GEMMs (fp32 16x16x4) ----------------
// One wave computes a 16x16 tile of out = act(A @ W + bias).
// A rows striped: lane l (l<16) holds row l, VGPR k-pair {0,1}; lanes 16-31
// hold k-pair {2,3}.  B symmetric with N in place of M.  C/D per ISA table.

// K=256 GEMM over concatenated inputs [A0 | A1*rowscale1], W:[256,128], ReLU.
__global__ void __launch_bounds__(32)
gemm_concat256_wmma(const float* __restrict__ A0, int lda0,
                    const float* __restrict__ A1, int lda1,
                    const float* __restrict__ rowscale1,   // nullable, scales A1 rows
                    const float* __restrict__ W,
                    const float* __restrict__ bias,
                    float* __restrict__ out) {
    const int lane = threadIdx.x & 31;
    const int half = lane >> 4;          // k-pair select
    const int l16  = lane & 15;
    const int rowA = blockIdx.x * 16 + l16;
    const int colB = blockIdx.y * 16 + l16;
    const float rs = rowscale1 ? rowscale1[rowA] : 1.0f;

    v8f acc = {};
    const float* a0p = A0 + (size_t)rowA * lda0 + half * 2;
    const float* a1p = A1 + (size_t)rowA * lda1 + half * 2;
    const float* bp0 = W + (size_t)(half * 2) * HDIM + colB;

    // first 128 K from A0
    for (int k0 = 0; k0 < 128; k0 += 4) {
        v2f a, b;
        a[0] = a0p[k0];
        a[1] = a0p[k0 + 1];
        const float* bp = bp0 + (size_t)k0 * HDIM;
        b[0] = bp[0];
        b[1] = bp[HDIM];
        acc = __builtin_amdgcn_wmma_f32_16x16x4_f32(false, a, false, b, (short)0, acc,
                                                    false, false);
    }
    // second 128 K from A1 (row-scaled)
    for (int k0 = 0; k0 < 128; k0 += 4) {
        v2f a, b;
        a[0] = a1p[k0] * rs;
        a[1] = a1p[k0 + 1] * rs;
        const float* bp = bp0 + (size_t)(k0 + 128) * HDIM;
        b[0] = bp[0];
        b[1] = bp[HDIM];
        acc = __builtin_amdgcn_wmma_f32_16x16x4_f32(false, a, false, b, (short)0, acc,
                                                    false, false);
    }

    const float bc = bias[colB];
    float* op = out + ((size_t)blockIdx.x * 16 + half * 8) * HDIM + colB;
    #pragma unroll
    for (int i = 0; i < 8; i++) {
        float v = acc[i] + bc;
        op[(size_t)i * HDIM] = fmaxf(v, 0.0f);      // ReLU
    }
}

// K=128 GraphConv GEMM: out = relu(BN((A*inv_in) @ W + bias)) [+ residual]
__global__ void __launch_bounds__(32)
gemm_gc128_wmma(const float* __restrict__ A,
                const float* __restrict__ inv_in,      // per-row scale (dst norm)
                const float* __restrict__ W,
                const float* __restrict__ bias,
                const float* __restrict__ gamma, const float* __restrict__ beta,
                const float* __restrict__ mean,  const float* __restrict__ var,
                const float* __restrict__ residual,    // nullable
                float* __restrict__ out) {
    const int lane = threadIdx.x & 31;
    const int half = lane >> 4;
    const int l16  = lane & 15;
    const int rowA = blockIdx.x * 16 + l16;
    const int colB = blockIdx.y * 16 + l16;
    const float rs = inv_in[rowA];

    v8f acc = {};
    const float* ap  = A + (size_t)rowA * HDIM + half * 2;
    const float* bp0 = W + (size_t)(half * 2) * HDIM + colB;
    for (int k0 = 0; k0 < 128; k0 += 4) {
        v2f a, b;
        a[0] = ap[k0] * rs;
        a[1] = ap[k0 + 1] * rs;
        const float* bp = bp0 + (size_t)k0 * HDIM;
        b[0] = bp[0];
        b[1] = bp[HDIM];
        acc = __builtin_amdgcn_wmma_f32_16x16x4_f32(false, a, false, b, (short)0, acc,
                                                    false, false);
    }

    // fused bias + BatchNorm(eval) + ReLU + residual epilogue (column = colB)
    const float sc = gamma[colB] * rsqrtf(var[colB] + BN_EPS);
    const float sh = beta[colB];
    const float mu = mean[colB];
    const float bc = bias[colB];
    const size_t baseRow = (size_t)blockIdx.x * 16 + half * 8;
    float* op = out + baseRow * HDIM + colB;
    const float* rp = residual ? residual + baseRow * HDIM + colB : nullptr;
    #pragma unroll
    for (int i = 0; i < 8; i++) {
        float v = (acc[i] + bc - mu) * sc + sh;
        v = fmaxf(v, 0.0f);
        if (rp) v += rp[(size_t)i * HDIM];
        op[(size_t)i * HDIM] = v;
    }
}

// Output head: [N,128] @ [128,C] + b.  One block per node; w reads coalesced,
// h row broadcast through L1.  0.64 GF -> not worth a WMMA pad path.
__global__ void out_head_kernel(const float* __restrict__ h, const float* __restrict__ W,
                                const float* __restrict__ b, float* __restrict__ out, int C) {
    int node = blockIdx.x;
    int c = threadIdx.x;
    if (c >= C) return;
    const float* hr = h + (size_t)node * HDIM;
    float s = b[c];
    #pragma unroll 8
    for (int k = 0; k < HDIM; k++) s += hr[k] * W[(size_t)k * C + c];
    out[(size_t)node * C + c] = s;
}

// ---------------- launcher ----------------

extern "C" void kernel_launch(void* const* d_in, const int* in_sizes, int n_in,
                              void* d_out, int out_size, void* d_ws, size_t ws_size,
                              hipStream_t stream) {
    const float* features = (const float*)d_in[0];
    const int*   src      = (const int*)d_in[1];
    const int*   dst      = (const int*)d_in[2];
    const float* w_in     = (const float*)d_in[3];
    const float* b_in     = (const float*)d_in[4];
    const float* gc_w     = (const float*)d_in[5];
    const float* gc_b     = (const float*)d_in[6];
    const float* bn_gamma = (const float*)d_in[7];
    const float* bn_beta  = (const float*)d_in[8];
    const float* bn_mean  = (const float*)d_in[9];
    const float* bn_var   = (const float*)d_in[10];
    const float* attn_w1  = (const float*)d_in[11];
    const float* attn_b1  = (const float*)d_in[12];
    const float* attn_w2  = (const float*)d_in[13];
    const float* attn_b2  = (const float*)d_in[14];
    const float* agg_w    = (const float*)d_in[15];
    const float* agg_b    = (const float*)d_in[16];
    const float* out_w    = (const float*)d_in[17];
    const float* out_b    = (const float*)d_in[18];

    const int IN = 256, H = HDIM;
    const int N = in_sizes[0] / IN;     // 50000
    const int E = in_sizes[1];          // 800000
    const int C = in_sizes[18];         // 50
    const int L = in_sizes[6] / H;      // 3

    // workspace carve-up (floats): 6 N-vectors + 3 N*H ping-pong buffers
    float* ws       = (float*)d_ws;
    float* in_deg   = ws;
    float* out_deg  = ws + (size_t)N;
    float* inv_in   = ws + 2 * (size_t)N;
    float* inv_out  = ws + 3 * (size_t)N;
    float* inv_mean = ws + 4 * (size_t)N;
    float* attn     = ws + 5 * (size_t)N;
    size_t NH = (size_t)N * H;
    float* B0 = ws + 6 * (size_t)N;   // h -> h_att -> layer ping
    float* B1 = B0 + NH;              // scatter accum / attn hidden
    float* B2 = B1 + NH;              // agg -> layer pong

    const int mtiles = N / 16;        // 3125 (N divisible by 16)
    dim3 ggrid(mtiles, H / 16);       // (3125, 8)
    dim3 gblock(32);                  // one wave per 16x16 tile

    // 1) degrees + norms
    zero_kernel<<<512, 256, 0, stream>>>(in_deg, 2 * (size_t)N);
    degree_kernel<<<(E + 255) / 256, 256, 0, stream>>>(src, dst, in_deg, out_deg, E);
    invdeg_kernel<<<(N + 255) / 256, 256, 0, stream>>>(in_deg, out_deg, inv_in, inv_out,
                                                       inv_mean, N);

    // 2) h = relu(features @ w_in + b_in)   (split features 256 -> 128|128)
    gemm_concat256_wmma<<<ggrid, gblock, 0, stream>>>(features, IN, features + 128, IN,
                                                      nullptr, w_in, b_in, B0);

    // 3) neigh = mean over in-edges of h[src]   (sum in B1, mean fused into GEMM)
    zero_kernel<<<2048, 256, 0, stream>>>(B1, NH);
    long long tot = (long long)E * 32;
    scatter_add_kernel<<<(unsigned)((tot + 255) / 256), 256, 0, stream>>>(
        B0, nullptr, src, dst, B1, tot);

    // 4) agg = relu([h | neigh] @ agg_w + agg_b)
    gemm_concat256_wmma<<<ggrid, gblock, 0, stream>>>(B0, H, B1, H, inv_mean,
                                                      agg_w, agg_b, B2);

    // 5) a = relu([h | agg] @ attn_w1 + attn_b1); attn = sigmoid(a @ w2 + b2)
    gemm_concat256_wmma<<<ggrid, gblock, 0, stream>>>(B0, H, B2, H, nullptr,
                                                      attn_w1, attn_b1, B1);
    attn_score_kernel<<<(N + 3) / 4, 128, 0, stream>>>(B1, attn_w2, attn_b2, attn, N);

    // 6) h_att = h + agg * attn   (in place on B0)
    gated_add_kernel<<<(unsigned)((NH + 255) / 256), 256, 0, stream>>>(B0, B2, attn, NH);

    // 7) L GraphConv layers: scatter (src-norm fused) then WMMA GEMM with
    //    dst-norm + bias + BN + ReLU + residual fused in the epilogue.
    float* cur = B0;
    float* nxt = B2;
    for (int i = 0; i < L; i++) {
        zero_kernel<<<2048, 256, 0, stream>>>(B1, NH);
        scatter_add_kernel<<<(unsigned)((tot + 255) / 256), 256, 0, stream>>>(
            cur, inv_out, src, dst, B1, tot);
        gemm_gc128_wmma<<<ggrid, gblock, 0, stream>>>(
            B1, inv_in, gc_w + (size_t)i * H * H, gc_b + (size_t)i * H,
            bn_gamma + (size_t)i * H, bn_beta + (size_t)i * H,
            bn_mean + (size_t)i * H, bn_var + (size_t)i * H,
            (i > 0) ? cur : nullptr, nxt);
        float* t = cur; cur = nxt; nxt = t;
    }

    // 8) out = h @ out_w + out_b   (cur holds final h)
    out_head_kernel<<<N, 64, 0, stream>>>(cur, out_w, out_b, (float*)d_out, C);
}